// Model_33062658245224
// MI455X (gfx1250) — compile-verified
//
#include <hip/hip_runtime.h>
#include <hip/hip_bf16.h>

typedef float v2f __attribute__((ext_vector_type(2)));
typedef float v8f __attribute__((ext_vector_type(8)));

#define P_DIM 128
#define EPS_F 0.05f
#define T_ITERS 4
#define LDS_STRIDE 132   // 128 + 4 pad -> conflict-free ds_load_b64 fragments

// ------------------------------------------------------------------ helpers
__global__ void k_fill(float* __restrict__ p, float v, int n) {
  int i = blockIdx.x * blockDim.x + threadIdx.x;
  if (i < n) p[i] = v;
}

// w1c_rowsum[j] = sum_k W1c[j][k]   (128 threads, one row each)
__global__ void k_rowsum(const float* __restrict__ W, float* __restrict__ out) {
  int j = threadIdx.x;
  float s = 0.f;
  #pragma unroll 8
  for (int k = 0; k < P_DIM; ++k) s += W[j * P_DIM + k];
  out[j] = s;
}

// ------------------------------------------------------------------ CSR build (once per call)
// Only edges whose source is unselected contribute (coef==0 otherwise),
// so filter them out here: halves the per-iteration gather work.
__global__ void k_hist(const int* __restrict__ esrc, const int* __restrict__ edst,
                       const unsigned char* __restrict__ sel,
                       int* __restrict__ deg, int n) {
  int e = blockIdx.x * blockDim.x + threadIdx.x;
  if (e >= n) return;
  if (!sel[esrc[e]]) atomicAdd(&deg[edst[e]], 1);
}

// single-block exclusive scan over deg[0..n-1] -> rowStart[0..n]
__global__ __launch_bounds__(1024) void k_scan(const int* __restrict__ deg,
                                               int* __restrict__ rowStart, int n) {
  __shared__ int buf[1024];
  __shared__ int carry;
  const int tid = threadIdx.x;
  if (tid == 0) carry = 0;
  __syncthreads();
  for (int base = 0; base < n; base += 1024) {
    const int i = base + tid;
    const int v = (i < n) ? deg[i] : 0;
    buf[tid] = v;
    __syncthreads();
    #pragma unroll
    for (int off = 1; off < 1024; off <<= 1) {
      int t = (tid >= off) ? buf[tid - off] : 0;
      __syncthreads();
      buf[tid] += t;
      __syncthreads();
    }
    const int incl = buf[tid];
    const int c = carry;
    if (i < n) rowStart[i] = c + incl - v;
    const int tot = buf[1023];
    __syncthreads();
    if (tid == 0) carry = c + tot;
    __syncthreads();
  }
  if (tid == 0) rowStart[n] = carry;
}

__global__ void k_fillcsr(const int* __restrict__ esrc, const int* __restrict__ edst,
                          const float* __restrict__ ew,
                          const unsigned char* __restrict__ sel,
                          int* __restrict__ cursor,
                          int* __restrict__ csrSrc, float* __restrict__ csrW, int n) {
  int e = blockIdx.x * blockDim.x + threadIdx.x;
  if (e >= n) return;
  const int src = esrc[e];
  if (sel[src]) return;
  const int p = atomicAdd(&cursor[edst[e]], 1);
  csrSrc[p] = src;
  csrW[p] = ew[e];
}

// ------------------------------------------------------------------ WMMA GEMM
// D = epilogue(A @ W^T), A:[N,128], W:[128,128], fp32.
// Block = 8 waves; A tile (16x128) staged once in LDS (padded, conflict-free),
// each wave owns one 16x16 output tile.  K loop: 32 x V_WMMA_F32_16X16X4_F32.
enum { GEMM_PLAIN = 0, GEMM_MU = 1, GEMM_RELU = 2 };

template <int MODE>
__global__ __launch_bounds__(256) void k_gemm128(
    const float* __restrict__ A, const float* __restrict__ W,
    float* __restrict__ D,
    const float* __restrict__ accS,   // MU: per-row scalar folded in at staging
    const float* __restrict__ IC,     // MU: IncurredCosts [N]
    const float* __restrict__ csum,   // MU: [N]
    const float* __restrict__ w1cRow) // MU: [128]
{
  __shared__ float As[16 * LDS_STRIDE];

  const int lane  = threadIdx.x & 31;
  const int wave  = threadIdx.x >> 5;   // nTile 0..7
  const int mTile = blockIdx.x;
  const int l15   = lane & 15;
  const int kh2   = (lane >> 4) * 2;    // 0 or 2

  // ---- stage A tile (16x128) into LDS; fold accS broadcast for MU mode
  for (int v = threadIdx.x; v < 16 * 32; v += 256) {
    const int r = v >> 5;
    const int c = (v & 31) * 4;
    float4 x = *(const float4*)(A + (size_t)(mTile * 16 + r) * P_DIM + c);
    if (MODE == GEMM_MU) {
      const float s = accS[mTile * 16 + r];
      x.x += s; x.y += s; x.z += s; x.w += s;
    }
    *(float4*)(&As[r * LDS_STRIDE + c]) = x;
  }
  __syncthreads();

  const int nB = wave * 16 + l15;       // W row (== B column)
  const float* __restrict__ Wrow = W + (size_t)nB * P_DIM + kh2;
  const float* __restrict__ Alds = &As[l15 * LDS_STRIDE + kh2];

  v8f c = {};
  #pragma unroll 4
  for (int k0 = 0; k0 < P_DIM; k0 += 4) {
    v2f a = *(const v2f*)(Alds + k0);
    v2f b = *(const v2f*)(Wrow + k0);
    c = __builtin_amdgcn_wmma_f32_16x16x4_f32(false, a, false, b,
                                              (short)0, c, false, false);
  }

  const int col = wave * 16 + l15;
  const int rowBase = mTile * 16 + (lane >> 4) * 8;
  #pragma unroll
  for (int i = 0; i < 8; ++i) {
    const int row = rowBase + i;
    float v = c[i];
    if (MODE == GEMM_PLAIN) {
      D[(size_t)row * P_DIM + col] = v;
    } else if (MODE == GEMM_RELU) {
      D[(size_t)row * P_DIM + col] = fmaxf(v, 0.f);
    } else { // GEMM_MU
      float extra = fmaxf(csum[row] * w1cRow[col], 0.f);
      D[(size_t)row * P_DIM + col] = IC[row] + fmaxf(v, 0.f) + extra;
    }
  }
}

// ------------------------------------------------------------------ CSR gather
// One wave per destination node: accumulate incoming messages in registers,
// single float4 store per lane.  Zero atomics; acc/accS fully written.
__global__ __launch_bounds__(256) void k_gather(
    const float* __restrict__ t2, const float* __restrict__ Ps,
    const int* __restrict__ rowStart, const int* __restrict__ csrSrc,
    const float* __restrict__ csrW, const int* __restrict__ steps,
    float* __restrict__ acc, float* __restrict__ accS, int n)
{
  const int lane = threadIdx.x & 31;
  const int node = blockIdx.x * 8 + (threadIdx.x >> 5);
  if (node >= n) return;
  const float S = (float)steps[0];
  const int jb = rowStart[node];
  const int je = rowStart[node + 1];

  float4 av = make_float4(0.f, 0.f, 0.f, 0.f);
  float as = 0.f;
  for (int j = jb; j < je; ++j) {
    const int src = csrSrc[j];                     // broadcast load
    const float coef = Ps[src] * S;                // broadcast load
    const float4 tv = *(const float4*)(t2 + (size_t)src * P_DIM + lane * 4);
    av.x += coef * tv.x; av.y += coef * tv.y;
    av.z += coef * tv.z; av.w += coef * tv.w;
    if (lane == 0) as += coef * csrW[j];
  }
  *(float4*)(acc + (size_t)node * P_DIM + lane * 4) = av;
  if (lane == 0) accS[node] = as;
}

// complex edges: scalar scatter, one thread per edge (only 0.4M atomics)
__global__ void k_cedges(
    const float* __restrict__ Ps, const float* __restrict__ cw,
    const int* __restrict__ csrc, const int* __restrict__ cdst,
    const unsigned char* __restrict__ selected,
    const int* __restrict__ steps,
    float* __restrict__ csum, int n)
{
  int e = blockIdx.x * blockDim.x + threadIdx.x;
  if (e >= n) return;
  const int src = csrc[e];
  const float w = cw[e];
  const float cc = selected[src] ? w : Ps[src] * (float)steps[0] * w;
  unsafeAtomicAdd(csum + cdst[e], cc);
}

// ------------------------------------------------------------------ Qs = clamp(h . WQ1)
__global__ __launch_bounds__(256) void k_qdot(
    const float* __restrict__ h, const float* __restrict__ WQ1,
    float* __restrict__ Qs, const int* __restrict__ hcc, int n)
{
  const int lane = threadIdx.x & 31;
  const int row = blockIdx.x * 8 + (threadIdx.x >> 5);
  if (row >= n) return;
  const float4 hv = *(const float4*)(h + (size_t)row * P_DIM + lane * 4);
  const float4 wv = *(const float4*)(WQ1 + lane * 4);
  float s = hv.x * wv.x + hv.y * wv.y + hv.z * wv.z + hv.w * wv.w;
  #pragma unroll
  for (int off = 16; off; off >>= 1) s += __shfl_xor(s, off, 32);
  if (lane == 0) Qs[row] = fmaxf(s, -(float)hcc[0]);
}

// ------------------------------------------------------------------ softmax -> Ps
__global__ __launch_bounds__(1024) void k_softmax(
    const float* __restrict__ Qs, float* __restrict__ Ps, int n)
{
  __shared__ float red[1024];
  __shared__ float gshared;
  const int tid = threadIdx.x;

  float m = -3.402823466e38f;
  for (int i = tid; i < n; i += 1024) m = fmaxf(m, Qs[i]);
  red[tid] = m; __syncthreads();
  for (int s = 512; s; s >>= 1) {
    if (tid < s) red[tid] = fmaxf(red[tid], red[tid + s]);
    __syncthreads();
  }
  if (tid == 0) gshared = red[0];
  __syncthreads();
  const float gmax = gshared;
  __syncthreads();

  float a = 0.f;
  for (int i = tid; i < n; i += 1024) a += __expf(Qs[i] - gmax);
  red[tid] = a; __syncthreads();
  for (int s = 512; s; s >>= 1) {
    if (tid < s) red[tid] += red[tid + s];
    __syncthreads();
  }
  if (tid == 0) gshared = red[0];
  __syncthreads();
  const float inv = (1.0f - EPS_F) / gshared;
  const float add = EPS_F / (float)n;
  for (int i = tid; i < n; i += 1024)
    Ps[i] = __expf(Qs[i] - gmax) * inv + add;
}

// ------------------------------------------------------------------ driver
extern "C" void kernel_launch(void* const* d_in, const int* in_sizes, int n_in,
                              void* d_out, int out_size, void* d_ws, size_t ws_size,
                              hipStream_t stream) {
  const float* edge_w  = (const float*)d_in[0];
  const float* cedge_w = (const float*)d_in[1];
  const float* IC      = (const float*)d_in[2];
  const float* W1      = (const float*)d_in[3];
  const float* W1c     = (const float*)d_in[4];
  const float* W2      = (const float*)d_in[5];
  const float* WQ1     = (const float*)d_in[6];
  const float* WQ2     = (const float*)d_in[7];
  const int* esrc      = (const int*)d_in[8];
  const int* edst      = (const int*)d_in[9];
  const int* csrc      = (const int*)d_in[10];
  const int* cdst      = (const int*)d_in[11];
  const unsigned char* sel = (const unsigned char*)d_in[12];
  const int* steps     = (const int*)d_in[13];
  const int* hcc       = (const int*)d_in[14];

  const int E  = in_sizes[0];
  const int E2 = in_sizes[1];
  const int N  = in_sizes[2];          // 50000, multiple of 16
  const size_t NP4 = (size_t)N * P_DIM * sizeof(float);

  // workspace carve-up (256B aligned)
  size_t off = 0;
  auto carve = [&](size_t bytes) {
    void* p = (char*)d_ws + off;
    off += (bytes + 255) & ~(size_t)255;
    return p;
  };
  float* mu       = (float*)carve(NP4);               // node embeddings
  float* bufA     = (float*)carve(NP4);               // t2, then h (reused)
  float* acc      = (float*)carve(NP4);               // gathered vector messages
  float* accS     = (float*)carve((size_t)N * 4);
  float* csum     = (float*)carve((size_t)N * 4);
  float* Ps       = (float*)carve((size_t)N * 4);
  float* w1cRow   = (float*)carve(P_DIM * 4);
  int*   deg      = (int*)carve((size_t)N * 4);
  int*   rowStart = (int*)carve((size_t)(N + 1) * 4);
  int*   cursor   = (int*)carve((size_t)N * 4);
  int*   csrSrc   = (int*)carve((size_t)E * 4);
  float* csrW     = (float*)carve((size_t)E * 4);

  float* Qs = (float*)d_out;

  const int gemmBlocks  = N / 16;       // 3125
  const int waveBlocksN = (N + 7) / 8;

  // ---- CSR by destination (edges & selected are call-invariant)
  hipMemsetAsync(deg, 0, (size_t)N * 4, stream);
  k_hist<<<(E + 255) / 256, 256, 0, stream>>>(esrc, edst, sel, deg, E);
  k_scan<<<1, 1024, 0, stream>>>(deg, rowStart, N);
  hipMemcpyAsync(cursor, rowStart, (size_t)N * 4, hipMemcpyDeviceToDevice, stream);
  k_fillcsr<<<(E + 255) / 256, 256, 0, stream>>>(esrc, edst, edge_w, sel,
                                                 cursor, csrSrc, csrW, E);

  // ---- state init
  hipMemsetAsync(mu, 0, NP4, stream);
  k_fill<<<(N + 255) / 256, 256, 0, stream>>>(Ps, 1.0f / (float)N, N);
  k_rowsum<<<1, P_DIM, 0, stream>>>(W1c, w1cRow);

  for (int it = 0; it < T_ITERS; ++it) {
    // t2 = mu @ W2^T
    k_gemm128<GEMM_PLAIN><<<gemmBlocks, 256, 0, stream>>>(
        mu, W2, bufA, nullptr, nullptr, nullptr, nullptr);

    hipMemsetAsync(csum, 0, (size_t)N * 4, stream);

    // acc/accS fully written by gather (no atomics, no memset)
    k_gather<<<waveBlocksN, 256, 0, stream>>>(
        bufA, Ps, rowStart, csrSrc, csrW, steps, acc, accS, N);
    k_cedges<<<(E2 + 255) / 256, 256, 0, stream>>>(
        Ps, cedge_w, csrc, cdst, sel, steps, csum, E2);

    // mu = IC + relu((acc + accS) @ W1^T) + relu(csum * w1c_rowsum)
    k_gemm128<GEMM_MU><<<gemmBlocks, 256, 0, stream>>>(
        acc, W1, mu, accS, IC, csum, w1cRow);

    // h = relu(mu @ WQ2^T)
    k_gemm128<GEMM_RELU><<<gemmBlocks, 256, 0, stream>>>(
        mu, WQ2, bufA, nullptr, nullptr, nullptr, nullptr);

    // Qs = max(h . WQ1, -hcc)  -> d_out
    k_qdot<<<waveBlocksN, 256, 0, stream>>>(bufA, WQ1, Qs, hcc, N);

    // Ps = softmax(Qs)*(1-eps) + eps/N  (dead after last iteration)
    if (it + 1 < T_ITERS)
      k_softmax<<<1, 1024, 0, stream>>>(Qs, Ps, N);
  }
}